// CV2DClassifier_32186484916321
// MI455X (gfx1250) — compile-verified
//
#include <hip/hip_runtime.h>

typedef float v2f __attribute__((ext_vector_type(2)));
typedef float v4f __attribute__((ext_vector_type(4)));
typedef float v8f __attribute__((ext_vector_type(8)));

#define N_MODES    64
#define N_CLASSES  10
#define TILE_ROWS  128
#define LDS_STRIDE 68   // 64 + 4 pad -> bank stride 4: conflict-free ds_load_b64 A-fragments

// ---------------------------------------------------------------------------
// Kernel 1: fold the whole affine pipeline into one [16x64] matrix + [16] const.
//   out[b,c] = sum_i x[b,i] * Mfold[c,i] + cst[c]
//   Mfold[c,i] = sum_m W[c,m] * S[2m, 2i]          (classes padded 10 -> 16 with 0)
//   cst[c]     = sum_m W[c,m] * (d[2m] + bias[m]) + b[c]
// ---------------------------------------------------------------------------
__global__ void cv2d_fold_kernel(const float* __restrict__ S, const float* __restrict__ d,
                                 const float* __restrict__ bias, const float* __restrict__ W,
                                 const float* __restrict__ b, float* __restrict__ Mfold,
                                 float* __restrict__ cst) {
  int t = blockIdx.x * blockDim.x + threadIdx.x;   // 0..1023
  if (t >= 16 * N_MODES) return;
  int c = t >> 6;        // padded class 0..15
  int i = t & 63;        // mode
  float acc = 0.f;
  if (c < N_CLASSES) {
    for (int m = 0; m < N_MODES; ++m)
      acc += W[c * N_MODES + m] * S[(2 * m) * (2 * N_MODES) + 2 * i];
  }
  Mfold[c * N_MODES + i] = acc;
  if (i == 0) {
    float cv = 0.f;
    if (c < N_CLASSES) {
      for (int m = 0; m < N_MODES; ++m)
        cv += W[c * N_MODES + m] * (d[2 * m] + bias[m]);
      cv += b[c];
    }
    cst[c] = cv;
  }
}

// ---------------------------------------------------------------------------
// Kernel 2: out[b, 0..9] = x[b, :] @ Mfold.T + cst   via V_WMMA_F32_16X16X4_F32.
// Block = 256 threads (8 waves), tile = 128 contiguous batch rows.
// ---------------------------------------------------------------------------
__global__ __launch_bounds__(256) void cv2d_gemm_kernel(const float* __restrict__ x,
                                                        const float* __restrict__ Mfold,
                                                        const float* __restrict__ cst,
                                                        float* __restrict__ out, int Bn) {
  __shared__ __align__(16) float lds[TILE_ROWS * LDS_STRIDE];

  const int t    = threadIdx.x;
  const int lane = t & 31;
  const int wave = t >> 5;
  const int r0   = blockIdx.x * TILE_ROWS;

  // --- Cooperative, fully-coalesced, non-temporal copy of the contiguous
  //     128x64 f32 x-tile into padded LDS. Thread t copies half-row (32 floats)
  //     of tile-row t/2. Branchless row clamp keeps EXEC full for the WMMAs.
  {
    const int row  = t >> 1;
    const int half = t & 1;
    int grow = r0 + row;
    if (grow >= Bn) grow = Bn - 1;
    const v4f* src = (const v4f*)(x + (size_t)grow * N_MODES + half * 32);
    float*     dst = lds + row * LDS_STRIDE + half * 32;
#pragma unroll
    for (int i = 0; i < 8; ++i) {
      v4f v = __builtin_nontemporal_load(src + i);     // global_load_b128 (TH=NT)
      *(v4f*)(dst + i * 4) = v;                        // ds_store_b128
    }
  }

  // --- Preload all B fragments (Mfold.T, K x 16) into registers.
  // B layout (f32 4x16): lane = N; v0 holds K = kc+0 (lanes 0-15) / kc+2 (lanes 16-31),
  //                      v1 holds K = kc+1 / kc+3.
  const int n     = lane & 15;               // class column
  const int khalf = (lane >> 4) << 1;        // 0 or 2 within each K-group of 4
  v2f bfrag[16];
#pragma unroll
  for (int tk = 0; tk < 16; ++tk)
    bfrag[tk] = *(const v2f*)(Mfold + n * N_MODES + tk * 4 + khalf);
  const float cn = cst[n];

  __syncthreads();

  // --- K-loop: 16 x V_WMMA_F32_16X16X4_F32, accumulator chained.
  // A layout (f32 16x4): lane = M (row, lanes 0-15 == 16-31); v0 = K+0/K+2, v1 = K+1/K+3.
  v8f acc = {};
  const float* arow = lds + (wave * 16 + (lane & 15)) * LDS_STRIDE + khalf;
#pragma unroll
  for (int tk = 0; tk < 16; ++tk) {
    v2f a = *(const v2f*)(arow + tk * 4);              // ds_load_2addr_b64, conflict-free
    acc = __builtin_amdgcn_wmma_f32_16x16x4_f32(
        /*neg_a=*/false, a, /*neg_b=*/false, bfrag[tk],
        /*c_mod=*/(short)0, acc, /*reuse_a=*/false, /*reuse_b=*/false);
  }

  // --- Store D: lane holds column n, VGPR v holds row m = v + 8*(lane>=16).
  // Hoist the row-bound check to a *scalar* whole-tile test: all blocks except
  // the single tail block take the branch-free path (one v_cmpx on n<10, then
  // 8 unconditional non-temporal global_store_b32).
  const int rbase = r0 + wave * 16 + ((lane >> 4) << 3);
  const bool tile_full = (r0 + TILE_ROWS) <= Bn;       // uniform -> s_cbranch
  if (n < N_CLASSES) {
    float* op = out + (size_t)rbase * N_CLASSES + n;
    if (tile_full) {
#pragma unroll
      for (int v = 0; v < 8; ++v)
        __builtin_nontemporal_store(acc[v] + cn, op + v * N_CLASSES);
    } else {
#pragma unroll
      for (int v = 0; v < 8; ++v)
        if (rbase + v < Bn)
          __builtin_nontemporal_store(acc[v] + cn, op + v * N_CLASSES);
    }
  }
}

extern "C" void kernel_launch(void* const* d_in, const int* in_sizes, int n_in,
                              void* d_out, int out_size, void* d_ws, size_t ws_size,
                              hipStream_t stream) {
  const float* x    = (const float*)d_in[0];   // [B, 64]
  const float* S    = (const float*)d_in[1];   // [128, 128]
  const float* d    = (const float*)d_in[2];   // [128]
  const float* bias = (const float*)d_in[3];   // [64]
  const float* W    = (const float*)d_in[4];   // [10, 64]
  const float* b    = (const float*)d_in[5];   // [10]
  float* out = (float*)d_out;                  // [B, 10]

  const int Bn = in_sizes[0] / N_MODES;

  float* Mfold = (float*)d_ws;                 // 16*64 floats
  float* cst   = Mfold + 16 * N_MODES;         // 16 floats

  hipLaunchKernelGGL(cv2d_fold_kernel, dim3(4), dim3(256), 0, stream,
                     S, d, bias, W, b, Mfold, cst);

  const int nblocks = (Bn + TILE_ROWS - 1) / TILE_ROWS;
  hipLaunchKernelGGL(cv2d_gemm_kernel, dim3(nblocks), dim3(256), 0, stream,
                     x, Mfold, cst, out, Bn);
}